// ChamferLoss_31044023616212
// MI455X (gfx1250) — compile-verified
//
#include <hip/hip_runtime.h>
#include <math.h>

typedef __attribute__((ext_vector_type(2))) float v2f;
typedef __attribute__((ext_vector_type(8))) float v8f;

#define N_PTS            8192
#define TILES_PER_BATCH  (N_PTS / 16)          // 512
#define N_TILES          (2 * TILES_PER_BATCH) // 1024
#define WAVES_PER_BLOCK  8

// Allow nnan/nsz fast-math locally: removes v_max v,v,v canonicalizations
// around v_min_num / v_max_num. Inputs are finite (normal(0,1) data).
#pragma float_control(precise, off, push)

// One wave per 16-point set1 tile. Sweeps all of set2 in 16-point chunks.
//
// The sq2 term is folded into the K dimension of the WMMA:
//   A row = (-2x1, -2y1, -2z1, 1)      (16x4 f32)
//   B col = ( x2,   y2,   z2,  sq2)    (4x16 f32)
//   D = A x B + 0 = sq2[col] - 2*cross[row,col]
// The row-constant sq1 term and the clamp-at-zero commute with min over
// columns, so they are applied once per wave after the sweep.
__global__ __launch_bounds__(256) void chamfer_tile_kernel(
    const float* __restrict__ set1, const float* __restrict__ set2,
    float* __restrict__ partial)
{
    const int lane = threadIdx.x & 31;
    const int wave = threadIdx.x >> 5;
    const int tile = blockIdx.x * WAVES_PER_BLOCK + wave;
    if (tile >= N_TILES) return;

    const int b  = tile / TILES_PER_BATCH;
    const int nt = tile % TILES_PER_BATCH;
    const float* __restrict__ s1 = set1 + (size_t)b * N_PTS * 3;
    const float* __restrict__ s2 = set2 + (size_t)b * N_PTS * 3;

    const int  col = lane & 15;
    const bool hi  = (lane >= 16);

    // ---- Loop-invariant A matrix: (K0,K1) on lanes 0-15, (K2,K3) on lanes 16-31.
    const int n_base = nt * 16;
    const float* p1 = s1 + (size_t)(n_base + col) * 3;
    const float x1 = p1[0], y1 = p1[1], z1 = p1[2];
    const float s1sq = x1 * x1 + y1 * y1 + z1 * z1;

    v2f a;
    a.x = hi ? (-2.0f * z1) : (-2.0f * x1);
    a.y = hi ? 1.0f         : (-2.0f * y1);

    // sq1 for the 8 rows this lane's C/D slots cover (rows v / v+8 per half).
    v8f sq1v;
#pragma unroll
    for (int v = 0; v < 8; ++v)
        sq1v[v] = __shfl(s1sq, (hi ? 8 : 0) + v, 32);

    v8f minv;   // running min over m of (sq2[col] - 2*cross)
#pragma unroll
    for (int v = 0; v < 8; ++v) minv[v] = 3.4e38f;

    const v8f czero = {};   // loop-invariant zero accumulator input

    // ---- Sweep set2: 512 chunks of 16 columns.
#pragma unroll 4
    for (int m = 0; m < N_PTS; m += 16) {
        const float* p2 = s2 + (size_t)(m + col) * 3;
        const float x2 = p2[0], y2 = p2[1], z2 = p2[2];
        const float s2sq = x2 * x2 + y2 * y2 + z2 * z2;

        v2f bb;
        bb.x = hi ? z2   : x2;
        bb.y = hi ? s2sq : y2;

        // D = A*B + 0  ->  sq2[col] - 2*cross tile
        v8f d = __builtin_amdgcn_wmma_f32_16x16x4_f32(
            /*neg_a=*/false, a, /*neg_b=*/false, bb,
            /*c_mod=*/(short)0, czero, /*reuse_a=*/false, /*reuse_b=*/false);

#pragma unroll
        for (int v = 0; v < 8; ++v)
            minv[v] = fminf(minv[v], d[v]);
    }

    // ---- Min over the 16 columns held across each 16-lane half (butterfly).
#pragma unroll
    for (int off = 1; off < 16; off <<= 1) {
#pragma unroll
        for (int v = 0; v < 8; ++v)
            minv[v] = fminf(minv[v], __shfl_xor(minv[v], off, 32));
    }

    // lanes 0-15 hold rows 0..7 (slot v), lanes 16-31 rows 8..15.
    // d2 = sq1[row] + min;  clamp, sqrt, sum.
    float psum = 0.0f;
#pragma unroll
    for (int v = 0; v < 8; ++v)
        psum += sqrtf(fmaxf(sq1v[v] + minv[v], 0.0f));
    psum += __shfl_xor(psum, 16, 32);   // combine the two halves

    if (lane == 0) partial[tile] = psum;
}

// Deterministic tree reduction of the 1024 per-tile partials.
__global__ __launch_bounds__(256) void chamfer_reduce_kernel(
    const float* __restrict__ partial, float* __restrict__ out)
{
    __shared__ float sm[256];
    float s = 0.0f;
    for (int i = threadIdx.x; i < N_TILES; i += 256) s += partial[i];
    sm[threadIdx.x] = s;
    __syncthreads();
#pragma unroll
    for (int stride = 128; stride > 0; stride >>= 1) {
        if ((int)threadIdx.x < stride) sm[threadIdx.x] += sm[threadIdx.x + stride];
        __syncthreads();
    }
    if (threadIdx.x == 0) out[0] = sm[0];
}

#pragma float_control(pop)

extern "C" void kernel_launch(void* const* d_in, const int* in_sizes, int n_in,
                              void* d_out, int out_size, void* d_ws, size_t ws_size,
                              hipStream_t stream)
{
    const float* set1 = (const float*)d_in[0];   // [2, 8192, 3] f32
    const float* set2 = (const float*)d_in[1];   // [2, 8192, 3] f32
    float* out     = (float*)d_out;              // scalar
    float* partial = (float*)d_ws;               // N_TILES floats

    const int blocks = N_TILES / WAVES_PER_BLOCK;  // 128
    chamfer_tile_kernel<<<blocks, 256, 0, stream>>>(set1, set2, partial);
    chamfer_reduce_kernel<<<1, 256, 0, stream>>>(partial, out);
}